// DenseCapsule_39591008534928
// MI455X (gfx1250) — compile-verified
//
#include <hip/hip_runtime.h>
#include <math.h>

// ---------------- problem constants (from reference) ----------------
#define BB      8
#define NSPAT   288
#define CHN     32
#define JC      24          // output capsules
#define MD      16          // pose dims (4x4)
#define NI      (NSPAT*CHN) // 9216 input capsules
#define EPSF    1e-7f
#define ITEMP   500.0f
#define TWO_PI  6.2831853071795864f

typedef __attribute__((ext_vector_type(2))) float v2f;
typedef __attribute__((ext_vector_type(8))) float v8f;

// stats layout per (b,j): [0]=sum r, [1..16]=sum r*v, [17..32]=sum r*v^2
#define STATS_STRIDE 33
#define STATS_FLOATS (BB*JC*STATS_STRIDE)   // 6336

__global__ __launch_bounds__(256) void caps_zero(float* __restrict__ p, int n) {
    int i = blockIdx.x * 256 + threadIdx.x;
    if (i < n) p[i] = 0.0f;
}

// One block = 8 waves; each wave handles one group = (ch, 4 consecutive n).
// Grid: x = 288 (2304 groups / 8 per block), y = 8 (batch).
__global__ __launch_bounds__(256) void caps_accum(
    const float* __restrict__ pose,     // (B, 288, 32, 16)
    const float* __restrict__ act,      // (B, 288, 32, 1)
    const float* __restrict__ w,        // (32, 24, 4, 4)
    const float* __restrict__ meanIn,   // (B, 24, 16)  valid when iter>0
    const float* __restrict__ varIn,    // (B, 24, 16)
    const float* __restrict__ ajIn,     // (B, 24)
    float* __restrict__ stats,          // (B, 24, 33)
    int iter)
{
    __shared__ float sVotes[8][4][JC][MD];   // 48 KB: per-wave vote tiles
    __shared__ float sStats[JC][STATS_STRIDE];
    __shared__ float sMean[JC][MD];
    __shared__ float sVar[JC][MD];
    __shared__ float sAj[JC];

    const int b    = blockIdx.y;
    const int tid  = threadIdx.x;
    const int wave = tid >> 5;
    const int lane = tid & 31;

    for (int x = tid; x < JC*STATS_STRIDE; x += 256) ((float*)sStats)[x] = 0.0f;
    if (iter > 0) {
        for (int x = tid; x < JC*MD; x += 256) {
            ((float*)sMean)[x] = meanIn[(size_t)b*JC*MD + x];
            ((float*)sVar)[x]  = varIn [(size_t)b*JC*MD + x];
        }
        if (tid < JC) sAj[tid] = ajIn[(size_t)b*JC + tid];
    }
    __syncthreads();

    // ---- group for this wave: same ch, four n's ----
    const int g  = blockIdx.x * 8 + wave;   // 0..2303
    const int ch = g / 72;
    const int n0 = (g % 72) * 4;

    const int l  = lane & 15;
    const int hi = lane >> 4;

    // B operand: 4x16 (K x N). N = l -> input u=l>>2, pose column c=l&3.
    // VGPR0 holds K = 2*hi, VGPR1 holds K = 2*hi+1 (row-striped, K split at lane 16).
    const int uB = l >> 2;
    const int cB = l & 3;
    const float* pBase = pose + (((size_t)b * NSPAT + (n0 + uB)) * CHN + ch) * MD;
    v2f bmat;
    bmat.x = pBase[(2*hi + 0) * 4 + cB];
    bmat.y = pBase[(2*hi + 1) * 4 + cB];

    // 6 WMMAs: each computes votes for 4 j's x 4 inputs (16x16 f32 tile)
    for (int jb = 0; jb < 6; ++jb) {
        // A operand: 16x4, row = stacked W row (j*4 + arow), lane = row,
        // VGPR0 = K 2*hi, VGPR1 = K 2*hi+1
        int row  = 16*jb + l;
        int j    = row >> 2;
        int arow = row & 3;
        const float* wBase = w + (((size_t)ch * JC + j) * 4 + arow) * 4;
        v2f amat;
        amat.x = wBase[2*hi + 0];
        amat.y = wBase[2*hi + 1];

        v8f dacc = {};
        dacc = __builtin_amdgcn_wmma_f32_16x16x4_f32(
            /*neg_a=*/false, amat, /*neg_b=*/false, bmat,
            /*c_mod=*/(short)0, dacc, /*reuse_a=*/false, /*reuse_b=*/false);

        // Scatter D: row m = vr + 8*hi, col = l. vote[u][j = jb*4 + m>>2][(m&3)*4 + c]
        #pragma unroll
        for (int vr = 0; vr < 8; ++vr) {
            int m = vr + 8*hi;
            sVotes[wave][uB][jb*4 + (m >> 2)][(m & 3)*4 + cB] = dacc[vr];
        }
    }
    __syncthreads();

    // ---- routing phase: lane -> input u2 = lane>>3, j's = (lane&7)+{0,8,16} ----
    const int u2 = lane >> 3;
    const int j0 = lane & 7;
    const int iGlob = (n0 + u2) * CHN + ch;
    float ai = act[(size_t)b * NI + iGlob];
    ai = (ai >= 0.0f) ? ai : 0.0f;   // jnp.where(acts >= ROUTE_MIN, ...)

    float rv[3];
    if (iter == 0) {
        float r0 = ai * (1.0f / (float)JC);
        rv[0] = r0; rv[1] = r0; rv[2] = r0;
    } else {
        float lap[3];
        #pragma unroll
        for (int t = 0; t < 3; ++t) {
            int j = j0 + 8*t;
            float lp = 0.0f;
            #pragma unroll
            for (int m = 0; m < MD; ++m) {
                float va = sVar[j][m] + EPSF;
                float dv = sVotes[wave][u2][j][m] - sMean[j][m];
                lp += __logf(TWO_PI * va) + dv*dv / va;
            }
            lap[t] = __logf(sAj[j] + EPSF) - 0.5f * lp;
        }
        // softmax over the 24 j's: 3 locals + xor-reduce over the 8-lane subgroup
        float mx = fmaxf(fmaxf(lap[0], lap[1]), lap[2]);
        #pragma unroll
        for (int s = 1; s <= 4; s <<= 1) mx = fmaxf(mx, __shfl_xor(mx, s, 32));
        float e0 = __expf(lap[0]-mx), e1 = __expf(lap[1]-mx), e2 = __expf(lap[2]-mx);
        float tot = e0 + e1 + e2;
        #pragma unroll
        for (int s = 1; s <= 4; s <<= 1) tot += __shfl_xor(tot, s, 32);
        float sc = ai / tot;
        rv[0] = e0*sc; rv[1] = e1*sc; rv[2] = e2*sc;
    }

    // accumulate: reduce across the 4 inputs of the wave (xor 8,16), then LDS atomics
    #pragma unroll
    for (int t = 0; t < 3; ++t) {
        int j = j0 + 8*t;
        float rs = rv[t];
        rs += __shfl_xor(rs, 8, 32);
        rs += __shfl_xor(rs, 16, 32);
        if (u2 == 0) atomicAdd(&sStats[j][0], rs);
        #pragma unroll
        for (int m = 0; m < MD; ++m) {
            float v  = sVotes[wave][u2][j][m];
            float s1 = rv[t] * v;
            float s2 = s1 * v;
            s1 += __shfl_xor(s1, 8, 32);  s1 += __shfl_xor(s1, 16, 32);
            s2 += __shfl_xor(s2, 8, 32);  s2 += __shfl_xor(s2, 16, 32);
            if (u2 == 0) {
                atomicAdd(&sStats[j][1 + m],  s1);
                atomicAdd(&sStats[j][17 + m], s2);
            }
        }
    }
    __syncthreads();

    // block-level partials -> global stats (L2-resident, 25 KB)
    for (int x = tid; x < JC*STATS_STRIDE; x += 256)
        atomicAdd(&stats[(size_t)b*JC*STATS_STRIDE + x], ((float*)sStats)[x]);
}

// m-step: 192 (b,j) pairs in one block.
__global__ __launch_bounds__(256) void caps_mstep(
    const float* __restrict__ stats,
    const float* __restrict__ beta_v,   // (24,16)
    const float* __restrict__ beta_a,   // (24,1)
    float* __restrict__ meanOut, float* __restrict__ varOut, float* __restrict__ ajOut,
    float* __restrict__ finalOut, int isFinal)
{
    int idx = threadIdx.x;
    if (idx >= BB*JC) return;
    int j = idx % JC;
    const float* S = stats + (size_t)idx * STATS_STRIDE;
    float den = S[0];
    float inv = 1.0f / (den + EPSF);
    float costSum = 0.0f;
    float mu[MD], va[MD];
    #pragma unroll
    for (int m = 0; m < MD; ++m) {
        float s1 = S[1 + m], s2 = S[17 + m];
        float mm = s1 * inv;
        // sum r (v-mu)^2 = s2 - 2 mu s1 + mu^2 den
        float vv = (s2 - 2.0f*mm*s1 + mm*mm*den) * inv;
        vv = fmaxf(vv, 0.0f);
        mu[m] = mm; va[m] = vv;
        costSum += beta_v[j*MD + m] + __logf(sqrtf(vv + EPSF) + EPSF);
    }
    float aj = 1.0f / (1.0f + __expf(-ITEMP * (beta_a[j] - costSum * den)));
    #pragma unroll
    for (int m = 0; m < MD; ++m) { meanOut[(size_t)idx*MD + m] = mu[m]; varOut[(size_t)idx*MD + m] = va[m]; }
    ajOut[idx] = aj;
    if (isFinal) {
        #pragma unroll
        for (int m = 0; m < MD; ++m) finalOut[(size_t)idx*MD + m] = mu[m];
        finalOut[BB*JC*MD + idx] = aj;   // a_j block after mean block
    }
}

extern "C" void kernel_launch(void* const* d_in, const int* in_sizes, int n_in,
                              void* d_out, int out_size, void* d_ws, size_t ws_size,
                              hipStream_t stream) {
    (void)in_sizes; (void)n_in; (void)out_size; (void)ws_size;
    const float* pose   = (const float*)d_in[0];
    const float* act    = (const float*)d_in[1];
    const float* w      = (const float*)d_in[2];
    const float* beta_v = (const float*)d_in[3];
    const float* beta_a = (const float*)d_in[4];
    float* out = (float*)d_out;

    float* ws    = (float*)d_ws;
    float* stats = ws;                               // 6336 floats
    float* meanB = ws + STATS_FLOATS;                // 3072
    float* varB  = meanB + BB*JC*MD;                 // 3072
    float* ajB   = varB  + BB*JC*MD;                 // 192

    for (int it = 0; it < 3; ++it) {
        caps_zero<<<(STATS_FLOATS + 255)/256, 256, 0, stream>>>(stats, STATS_FLOATS);
        caps_accum<<<dim3(288, BB), 256, 0, stream>>>(pose, act, w, meanB, varB, ajB, stats, it);
        caps_mstep<<<1, 256, 0, stream>>>(stats, beta_v, beta_a, meanB, varB, ajB,
                                          out, (it == 2) ? 1 : 0);
    }
}